// LocalGMMScorerAttention_70033736728792
// MI455X (gfx1250) — compile-verified
//
#include <hip/hip_runtime.h>
#include <hip/hip_bf16.h>
#include <math.h>

#define B_   32
#define L_   2048
#define DCTX 512
#define DQ   1024
#define H_   256
#define WIN  7
#define PL   (-3)
#define PITCH 516   // LDS row pitch (floats): 516 mod 64 banks = 4 -> conflict-free frags

typedef __attribute__((ext_vector_type(2))) float v2f;
typedef __attribute__((ext_vector_type(8))) float v8f;

// Workspace layout (floats):
//   [0,        B_*H_)      h      = tanh(query @ Wq + bq)          [32,256]
//   [B_*H_,  2*B_*H_)      qW1    = query @ W1[512:,:]             [32,256]
//   [2*B_*H_, +B_*4)       stats: per b {alpha, beta, kappa, center}

// ---------------------------------------------------------------------------
// Kernel 1: [32,1024] @ [1024,256] twice (Wq and the query-half of W1),
// fp32 WMMA 16x16x4, one wave per 16x16 output tile. Weight rows prefetched.
// ---------------------------------------------------------------------------
__global__ __launch_bounds__(32)
void k1_qproj(const float* __restrict__ query,
              const float* __restrict__ Wq,
              const float* __restrict__ bq,
              const float* __restrict__ W1,
              float* __restrict__ ws)
{
  const int lane  = threadIdx.x;
  const int n0    = blockIdx.x * 16;
  const int m0    = blockIdx.y * 16;
  const bool isH  = (blockIdx.z == 0);
  const float* Bp = isH ? Wq : (W1 + (size_t)DCTX * H_); // row-major [1024,256]
  float* outp     = isH ? ws : (ws + B_ * H_);

  const int mrow  = lane & 15;
  const int khalf = (lane >> 4) << 1;        // lanes 0-15 -> K+{0,1}, 16-31 -> K+{2,3}
  const int ncol  = n0 + (lane & 15);
  const float* arow = query + (size_t)(m0 + mrow) * DQ;

  v8f acc = {};
  for (int k0 = 0; k0 < DQ; k0 += 32) {
    // stream-prefetch the weight rows ~3 chunks ahead (speculative, clamped)
    const int kp = (k0 + 96 < DQ) ? (k0 + 96) : (DQ - 1);
    __builtin_prefetch(&Bp[(size_t)kp * H_ + ncol], 0, 1);
    #pragma unroll
    for (int k = k0; k < k0 + 32; k += 4) {
      v2f a;
      a.x = arow[k + khalf];
      a.y = arow[k + khalf + 1];
      v2f b;
      b.x = Bp[(size_t)(k + khalf) * H_ + ncol];
      b.y = Bp[(size_t)(k + khalf + 1) * H_ + ncol];
      acc = __builtin_amdgcn_wmma_f32_16x16x4_f32(false, a, false, b,
                                                  (short)0, acc, false, false);
    }
  }

  const int rbase = (lane < 16) ? 0 : 8;     // C/D layout: VGPR r = rows r / r+8
  #pragma unroll
  for (int r = 0; r < 8; ++r) {
    const int mm = m0 + rbase + r;
    float v = acc[r];
    if (isH) v = tanhf(v + bq[ncol]);
    outp[(size_t)mm * H_ + ncol] = v;
  }
}

// ---------------------------------------------------------------------------
// Kernel 2: per-batch stats. stat = h @ Ws + bs -> alpha, beta, kappa, center.
// One wave per batch row; shuffle reduction.
// ---------------------------------------------------------------------------
__global__ __launch_bounds__(32)
void k2_stats(const float* __restrict__ kappa_prev,
              const float* __restrict__ Ws,
              const float* __restrict__ bs,
              float* __restrict__ ws)
{
  const int b    = blockIdx.x;
  const int lane = threadIdx.x;
  const float* h = ws + (size_t)b * H_;

  float s0 = 0.f, s1 = 0.f, s2 = 0.f;
  for (int j = lane; j < H_; j += 32) {
    const float hv = h[j];
    s0 += hv * Ws[j * 3 + 0];
    s1 += hv * Ws[j * 3 + 1];
    s2 += hv * Ws[j * 3 + 2];
  }
  #pragma unroll
  for (int off = 16; off > 0; off >>= 1) {
    s0 += __shfl_xor(s0, off, 32);
    s1 += __shfl_xor(s1, off, 32);
    s2 += __shfl_xor(s2, off, 32);
  }
  if (lane == 0) {
    const float alpha = expf(s0 + bs[0]);
    const float beta  = expf(s1 + bs[1]);
    const float kappa = expf(s2 + bs[2]) + kappa_prev[b];
    float* st = ws + 2 * B_ * H_ + b * 4;
    st[0] = alpha; st[1] = beta; st[2] = kappa;
    st[3] = rintf(kappa);                    // nearest-even, matches jnp.round
  }
}

// ---------------------------------------------------------------------------
// Kernel 3: per-batch window scorer + normalization + outputs.
// Block = 512 threads = 16 waves; wave w owns hidden-unit tile [16w, 16w+16).
// The <=7-row ctx window is staged ONCE into LDS (33 KB of 320 KB/WGP) and
// feeds both the WMMA A-fragments (ds_load_b64) and the expected_ctx sum,
// instead of 16 redundant global streams.
// ---------------------------------------------------------------------------
__global__ __launch_bounds__(512)
void k3_window(const float* __restrict__ ctx,
               const float* __restrict__ W1,
               const float* __restrict__ b1,
               const float* __restrict__ W2,
               const float* __restrict__ b2,
               const float* __restrict__ ws,
               float* __restrict__ out_expected,   // [32,512]
               float* __restrict__ out_pctx)       // [32,2048]
{
  __shared__ float s_ctx[16 * PITCH];
  __shared__ float s_score[16];
  __shared__ float s_pw[WIN];

  const int b    = blockIdx.x;
  const int tid  = threadIdx.x;
  const int lane = tid & 31;
  const int wave = tid >> 5;                 // 0..15 = hidden n-tile

  const float* st  = ws + 2 * B_ * H_ + b * 4;
  const float alpha = st[0], beta = st[1], kappa = st[2];
  const int   c     = (int)st[3];

  if (tid < 16) s_score[tid] = 0.f;

  // Stage the (clamped) 16 x 512 window into LDS with b128 loads/stores.
  {
    const float4* ctx4 = (const float4*)ctx;   // [B, L, 128] float4
    for (int i = tid; i < 16 * 128; i += 512) {
      const int row  = i >> 7;
      const int col4 = i & 127;
      int p = c + PL + row;
      p = min(max(p, 0), L_ - 1);              // invalid rows masked later
      const float4 v = ctx4[((size_t)b * L_ + p) * 128 + col4];
      *reinterpret_cast<float4*>(&s_ctx[row * PITCH + col4 * 4]) = v;
    }
  }
  __syncthreads();

  const int n0    = wave * 16;
  const int ncol  = n0 + (lane & 15);
  const int mrow  = lane & 15;
  const int khalf = (lane >> 4) << 1;
  const float* qrow = ws + B_ * H_ + (size_t)b * H_;
  const float* srow = &s_ctx[mrow * PITCH];

  v8f acc = {};
  for (int k0 = 0; k0 < DCTX; k0 += 32) {
    const int kp = (k0 + 96 < DCTX) ? (k0 + 96) : (DCTX - 1);
    __builtin_prefetch(&W1[(size_t)kp * H_ + ncol], 0, 1);
    #pragma unroll
    for (int k = k0; k < k0 + 32; k += 4) {
      v2f a;
      a.x = srow[k + khalf];                   // ds_load_b64, bank-conflict-free
      a.y = srow[k + khalf + 1];
      v2f bb;
      bb.x = W1[(size_t)(k + khalf) * H_ + ncol];
      bb.y = W1[(size_t)(k + khalf + 1) * H_ + ncol];
      acc = __builtin_amdgcn_wmma_f32_16x16x4_f32(false, a, false, bb,
                                                  (short)0, acc, false, false);
    }
  }

  // Fuse: u = tanh(ctx@W1c + q@W1q + b1); partial score = u * W2, reduced over
  // this wave's 16 hidden columns via half-wave shuffles.
  const float qadd = qrow[ncol] + b1[ncol];
  const float w2   = W2[ncol];
  float rp[8];
  #pragma unroll
  for (int r = 0; r < 8; ++r)
    rp[r] = tanhf(acc[r] + qadd) * w2;
  #pragma unroll
  for (int off = 1; off < 16; off <<= 1) {
    #pragma unroll
    for (int r = 0; r < 8; ++r)
      rp[r] += __shfl_xor(rp[r], off, 32);
  }
  if (lane == 0) {
    #pragma unroll
    for (int r = 0; r < 8; ++r) atomicAdd(&s_score[r], rp[r]);       // rows 0..7
  } else if (lane == 16) {
    #pragma unroll
    for (int r = 0; r < 8; ++r) atomicAdd(&s_score[8 + r], rp[r]);   // rows 8..15
  }
  __syncthreads();

  if (tid == 0) {
    float post[WIN];
    float sum = 0.f;
    #pragma unroll
    for (int m = 0; m < WIN; ++m) {
      const int pp = c + PL + m;
      float v = 0.f;
      if (pp >= 0 && pp < L_) {
        const float d     = (float)pp - kappa;
        const float prior = alpha * expf(-beta * d * d);
        const float like  = expf(s_score[m] + b2[0]);
        v = like * prior;
      }
      post[m] = v;
      sum += v;
    }
    const float inv = 1.f / sum;             // (sum==0 -> NaN, same as reference)
    #pragma unroll
    for (int m = 0; m < WIN; ++m) s_pw[m] = post[m] * inv;
  }
  __syncthreads();

  // p_ctx row: zero everywhere, pw inside the window.
  for (int idx = tid; idx < L_; idx += 512) {
    const int m = idx - (c + PL);
    out_pctx[(size_t)b * L_ + idx] = (m >= 0 && m < WIN) ? s_pw[m] : 0.f;
  }
  // expected_ctx from the staged LDS window (valid rows == staged rows).
  {
    const int d = tid;
    if (d < DCTX) {
      float e = 0.f;
      #pragma unroll
      for (int m = 0; m < WIN; ++m) {
        const int pp = c + PL + m;
        if (pp >= 0 && pp < L_)
          e += s_pw[m] * s_ctx[m * PITCH + d];
      }
      out_expected[(size_t)b * DCTX + d] = e;
    }
  }
}

// ---------------------------------------------------------------------------
extern "C" void kernel_launch(void* const* d_in, const int* in_sizes, int n_in,
                              void* d_out, int out_size, void* d_ws, size_t ws_size,
                              hipStream_t stream) {
  const float* query      = (const float*)d_in[0];
  const float* ctx        = (const float*)d_in[1];
  const float* kappa_prev = (const float*)d_in[2];
  const float* Wq         = (const float*)d_in[3];
  const float* bq         = (const float*)d_in[4];
  const float* Ws         = (const float*)d_in[5];
  const float* bs         = (const float*)d_in[6];
  const float* W1         = (const float*)d_in[7];
  const float* b1         = (const float*)d_in[8];
  const float* W2         = (const float*)d_in[9];
  const float* b2         = (const float*)d_in[10];

  float* ws           = (float*)d_ws;
  float* out_expected = (float*)d_out;              // [32,512]
  float* out_pctx     = out_expected + B_ * DCTX;   // [32,2048]

  dim3 g1(H_ / 16, B_ / 16, 2);                     // 16 n-tiles x 2 m-tiles x 2 outputs
  k1_qproj<<<g1, 32, 0, stream>>>(query, Wq, bq, W1, ws);
  k2_stats<<<B_, 32, 0, stream>>>(kappa_prev, Ws, bs, ws);
  k3_window<<<B_, 512, 0, stream>>>(ctx, W1, b1, W2, b2, ws, out_expected, out_pctx);
}